// SAGE_7584912244792
// MI455X (gfx1250) — compile-verified
//
#include <hip/hip_runtime.h>
#include <math.h>

#define N_NODES 100000
#define N_EDGES 1600000
#define IN_C    128
#define HID_C   256
#define OUT_C   47

typedef float v2f __attribute__((ext_vector_type(2)));
typedef float v8f __attribute__((ext_vector_type(8)));

// ---------------- workspace layout (floats) ----------------
// cnt/inv : [N_NODES]
// agg     : [N_NODES * HID_C]   (layer1 uses first N_NODES*IN_C floats, row stride IN_C)
// h       : [N_NODES * HID_C]
#define AGG_OFF 100096u                       // N_NODES rounded up to 256
#define H_OFF   (AGG_OFF + (size_t)N_NODES * HID_C)

// ---------------- utility kernels ----------------
__global__ void zero_f32_kernel(float* __restrict__ p, size_t n) {
    size_t i = (size_t)blockIdx.x * blockDim.x + threadIdx.x;
    size_t stride = (size_t)gridDim.x * blockDim.x;
    for (; i < n; i += stride) p[i] = 0.0f;
}

__global__ void count_kernel(const int* __restrict__ dst,
                             float* __restrict__ cnt, int nE) {
    int e = blockIdx.x * blockDim.x + threadIdx.x;
    if (e < nE) atomicAdd(&cnt[dst[e]], 1.0f);
}

__global__ void inv_kernel(float* __restrict__ cnt, int n) {
    int i = blockIdx.x * blockDim.x + threadIdx.x;
    if (i < n) cnt[i] = 1.0f / fmaxf(cnt[i], 1.0f);
}

// ---------------- scatter-add (segment_sum of gathered rows) ----------------
template<int C>
__global__ void scatter_kernel(const float4* __restrict__ feat,
                               const int* __restrict__ src,
                               const int* __restrict__ dst,
                               float* __restrict__ agg, int nE) {
    constexpr int G = C / 4;
    int tid = blockIdx.x * blockDim.x + threadIdx.x;
    int e = tid / G;
    int q = tid - e * G;
    if (e >= nE) return;
    int s = src[e], d = dst[e];
    float4 v = feat[(size_t)s * G + q];
    float* base = agg + (size_t)d * C + (size_t)q * 4;
    atomicAdd(base + 0, v.x);
    atomicAdd(base + 1, v.y);
    atomicAdd(base + 2, v.z);
    atomicAdd(base + 3, v.w);
}

// ---------------- layer 1: h = relu(mean @ W1_l + b1 + x @ W1_r) ----------------
// One wave per 16x16 tile; 4 waves per block cover 64 output cols.
__global__ __launch_bounds__(128)
void gemm1_kernel(const float* __restrict__ agg, const float* __restrict__ inv,
                  const float* __restrict__ x,   const float* __restrict__ Wl,
                  const float* __restrict__ b,   const float* __restrict__ Wr,
                  float* __restrict__ h) {
    const int lane    = threadIdx.x;                         // 0..31
    const int colbase = (blockIdx.x * 4 + threadIdx.y) * 16; // 0..240
    const int rowbase = blockIdx.y * 16;
    const int idx     = lane & 15;
    const int khalf   = lane >> 4;
    const int koff    = khalf * 2;                           // K offsets {0,1}|{2,3}
    const int row     = rowbase + idx;

    const float invr  = inv[row];
    const float* aggp = agg + (size_t)row * IN_C + koff;
    const float* xp   = x   + (size_t)row * IN_C + koff;

    v8f acc = {};
    #pragma unroll 4
    for (int kb = 0; kb < IN_C; kb += 4) {
        v2f am, ax, bl, br;
        am.x = aggp[kb]     * invr;
        am.y = aggp[kb + 1] * invr;
        ax.x = xp[kb];
        ax.y = xp[kb + 1];
        const float* wl = Wl + (size_t)(kb + koff) * HID_C + colbase + idx;
        const float* wr = Wr + (size_t)(kb + koff) * HID_C + colbase + idx;
        bl.x = wl[0]; bl.y = wl[HID_C];
        br.x = wr[0]; br.y = wr[HID_C];
        acc = __builtin_amdgcn_wmma_f32_16x16x4_f32(false, am, false, bl,
                                                    (short)0, acc, false, false);
        acc = __builtin_amdgcn_wmma_f32_16x16x4_f32(false, ax, false, br,
                                                    (short)0, acc, false, false);
    }

    const float bias = b[colbase + idx];
    const int r0 = rowbase + khalf * 8;                       // C/D row layout
    float* hp = h + (size_t)r0 * HID_C + colbase + idx;
    #pragma unroll
    for (int g = 0; g < 8; ++g) {
        float v = acc[g] + bias;
        hp[(size_t)g * HID_C] = fmaxf(v, 0.0f);               // fused ReLU
    }
}

// ---------------- layer 2 + log_softmax ----------------
// One wave owns a 16-row block and all 47 (padded 48) output columns.
__global__ __launch_bounds__(32)
void gemm2_kernel(const float* __restrict__ agg, const float* __restrict__ inv,
                  const float* __restrict__ h,   const float* __restrict__ Wl,
                  const float* __restrict__ b,   const float* __restrict__ Wr,
                  float* __restrict__ out) {
    const int lane    = threadIdx.x;
    const int rowbase = blockIdx.x * 16;
    const int idx     = lane & 15;
    const int khalf   = lane >> 4;
    const int koff    = khalf * 2;
    const int row     = rowbase + idx;

    const float invr  = inv[row];
    const float* aggp = agg + (size_t)row * HID_C + koff;
    const float* hp   = h   + (size_t)row * HID_C + koff;

    const int  c0 = idx, c1 = 16 + idx;
    const bool v2ok = (32 + idx) < OUT_C;
    const int  c2 = v2ok ? (32 + idx) : (OUT_C - 1);          // clamp, mask later

    v8f acc0 = {}, acc1 = {}, acc2 = {};
    #pragma unroll 4
    for (int kb = 0; kb < HID_C; kb += 4) {
        v2f am, ah;
        am.x = aggp[kb]     * invr;
        am.y = aggp[kb + 1] * invr;
        ah.x = hp[kb];
        ah.y = hp[kb + 1];
        const float* wl = Wl + (size_t)(kb + koff) * OUT_C;
        const float* wr = Wr + (size_t)(kb + koff) * OUT_C;
        v2f bl0 = {wl[c0], wl[OUT_C + c0]};
        v2f bl1 = {wl[c1], wl[OUT_C + c1]};
        v2f bl2 = {wl[c2], wl[OUT_C + c2]};
        v2f br0 = {wr[c0], wr[OUT_C + c0]};
        v2f br1 = {wr[c1], wr[OUT_C + c1]};
        v2f br2 = {wr[c2], wr[OUT_C + c2]};
        acc0 = __builtin_amdgcn_wmma_f32_16x16x4_f32(false, am, false, bl0, (short)0, acc0, false, false);
        acc0 = __builtin_amdgcn_wmma_f32_16x16x4_f32(false, ah, false, br0, (short)0, acc0, false, false);
        acc1 = __builtin_amdgcn_wmma_f32_16x16x4_f32(false, am, false, bl1, (short)0, acc1, false, false);
        acc1 = __builtin_amdgcn_wmma_f32_16x16x4_f32(false, ah, false, br1, (short)0, acc1, false, false);
        acc2 = __builtin_amdgcn_wmma_f32_16x16x4_f32(false, am, false, bl2, (short)0, acc2, false, false);
        acc2 = __builtin_amdgcn_wmma_f32_16x16x4_f32(false, ah, false, br2, (short)0, acc2, false, false);
    }

    const float bias0 = b[c0], bias1 = b[c1], bias2 = b[c2];
    const int r0 = rowbase + khalf * 8;

    #pragma unroll
    for (int g = 0; g < 8; ++g) {
        float x0 = acc0[g] + bias0;
        float x1 = acc1[g] + bias1;
        float x2 = acc2[g] + bias2;

        // row max across 16 lanes of this half (xor masks 1..8 keep halves apart)
        float m = fmaxf(x0, x1);
        m = v2ok ? fmaxf(m, x2) : m;
        #pragma unroll
        for (int d = 1; d < 16; d <<= 1) m = fmaxf(m, __shfl_xor(m, d, 32));

        float s = __expf(x0 - m) + __expf(x1 - m) + (v2ok ? __expf(x2 - m) : 0.0f);
        #pragma unroll
        for (int d = 1; d < 16; d <<= 1) s += __shfl_xor(s, d, 32);

        const float lse = m + __logf(s);
        float* op = out + (size_t)(r0 + g) * OUT_C;
        op[c0] = x0 - lse;
        op[c1] = x1 - lse;
        if (v2ok) op[32 + idx] = x2 - lse;
    }
}

// ---------------- launch ----------------
extern "C" void kernel_launch(void* const* d_in, const int* in_sizes, int n_in,
                              void* d_out, int out_size, void* d_ws, size_t ws_size,
                              hipStream_t stream) {
    (void)in_sizes; (void)n_in; (void)out_size; (void)ws_size;

    const float* x   = (const float*)d_in[0];
    const int*   ei  = (const int*)  d_in[1];
    const float* W1l = (const float*)d_in[2];
    const float* b1  = (const float*)d_in[3];
    const float* W1r = (const float*)d_in[4];
    const float* W2l = (const float*)d_in[5];
    const float* b2  = (const float*)d_in[6];
    const float* W2r = (const float*)d_in[7];

    const int* src = ei;
    const int* dst = ei + N_EDGES;

    float* ws  = (float*)d_ws;
    float* cnt = ws;                 // becomes inv-count after inv_kernel
    float* agg = ws + AGG_OFF;
    float* h   = ws + H_OFF;
    float* out = (float*)d_out;

    // ---- layer 1 ----
    zero_f32_kernel<<<4096, 256, 0, stream>>>(ws, AGG_OFF + (size_t)N_NODES * IN_C);
    count_kernel<<<(N_EDGES + 255) / 256, 256, 0, stream>>>(dst, cnt, N_EDGES);
    inv_kernel<<<(N_NODES + 255) / 256, 256, 0, stream>>>(cnt, N_NODES);

    {
        long long nthreads = (long long)N_EDGES * (IN_C / 4);
        int blocks = (int)((nthreads + 255) / 256);
        scatter_kernel<IN_C><<<blocks, 256, 0, stream>>>(
            (const float4*)x, src, dst, agg, N_EDGES);
    }

    gemm1_kernel<<<dim3(HID_C / 64, N_NODES / 16), dim3(32, 4), 0, stream>>>(
        agg, cnt, x, W1l, b1, W1r, h);

    // ---- layer 2 ----
    zero_f32_kernel<<<8192, 256, 0, stream>>>(agg, (size_t)N_NODES * HID_C);
    {
        long long nthreads = (long long)N_EDGES * (HID_C / 4);
        int blocks = (int)((nthreads + 255) / 256);
        scatter_kernel<HID_C><<<blocks, 256, 0, stream>>>(
            (const float4*)h, src, dst, agg, N_EDGES);
    }

    gemm2_kernel<<<N_NODES / 16, 32, 0, stream>>>(agg, cnt, h, W2l, b2, W2r, out);
}